// SegmentClassifier_41832981463622
// MI455X (gfx1250) — compile-verified
//
#include <hip/hip_runtime.h>
#include <math.h>

// ---------------- problem constants ----------------
#define B_     16
#define N_     1024
#define E_     2048
#define DIN    2
#define DH     8
#define DF     10   // D_IN + D_H
#define HS     16   // padded per-node feature stride (floats)
#define MS     32   // per-node message-vector stride: [mi(10)|mo(10)|H(10)|0,0]
#define NSPLIT 4    // N-dimension split for the decode pass

#define NCOLS  (B_ * E_)   // 32768 edge columns per incidence matrix

typedef float v2f __attribute__((ext_vector_type(2)));
typedef float v4f __attribute__((ext_vector_type(4)));
typedef float v8f __attribute__((ext_vector_type(8)));

// Branch-free activations using the hardware transcendentals
// (v_exp_f32 + v_rcp_f32; no exec-mask divergence, saturates correctly).
__device__ __forceinline__ float fast_tanh(float x) {
    float e = __expf(2.0f * x);                       // inf for large x -> 1
    return 1.0f - 2.0f * __builtin_amdgcn_rcpf(e + 1.0f);
}
__device__ __forceinline__ float fast_sigmoid(float z) {
    return __builtin_amdgcn_rcpf(1.0f + __expf(-z));
}

// =====================================================================
// 1a) Decode partial pass. One launch covers BOTH incidence matrices
//     (grid.z) and splits the N dimension 4 ways (grid.y) for memory-
//     level parallelism on the 256 MB stream. Nontemporal float4 loads
//     (don't thrash the 192 MB L2), branch-free prefetch every 8 rows.
// =====================================================================
__global__ void decode_partial_kernel(const float* __restrict__ Ri,
                                      const float* __restrict__ Ro,
                                      float* __restrict__ partial) {
    const float* __restrict__ R = blockIdx.z ? Ro : Ri;
    const int t  = blockIdx.x * blockDim.x + threadIdx.x;  // column group
    const int c4 = t * 4;
    const int b  = c4 / E_;
    const int e  = c4 % E_;
    const int n0 = blockIdx.y * (N_ / NSPLIT);
    const float* col = R + (size_t)b * N_ * E_ + e;

    v4f acc = {};
    for (int n = n0; n < n0 + N_ / NSPLIT; n += 8) {
        __builtin_prefetch(col + (size_t)(n + 16) * E_, 0, 1);  // global_prefetch_b8
#pragma unroll
        for (int k = 0; k < 8; ++k) {
            v4f v = __builtin_nontemporal_load(
                        (const v4f*)(col + (size_t)(n + k) * E_));
            acc += v * (float)(n + k);
        }
    }
    float* p = partial + ((size_t)blockIdx.z * NSPLIT + blockIdx.y) * NCOLS + c4;
    *(v4f*)p = acc;
}

// 1b) Combine the NSPLIT partials -> integer node indices.
__global__ void decode_combine_kernel(const float* __restrict__ partial,
                                      int* __restrict__ idx_i,
                                      int* __restrict__ idx_o) {
    const int c = blockIdx.x * blockDim.x + threadIdx.x;  // 0 .. 2*NCOLS
    const int z   = c / NCOLS;
    const int col = c % NCOLS;
    float s = 0.0f;
#pragma unroll
    for (int y = 0; y < NSPLIT; ++y)
        s += partial[((size_t)z * NSPLIT + y) * NCOLS + col];
    int* out = z ? idx_o : idx_i;
    out[col] = (int)(s + 0.5f);
}

// =====================================================================
// 2) Input encoder: H[node] = [tanh(Wi @ x + bi), x]
// =====================================================================
__global__ void encoder_kernel(const float* __restrict__ X,
                               const float* __restrict__ Wi,
                               const float* __restrict__ bi,
                               float* __restrict__ H) {
    int node = blockIdx.x * blockDim.x + threadIdx.x;
    if (node >= B_ * N_) return;
    float x0 = X[node * DIN + 0];
    float x1 = X[node * DIN + 1];
    float* h = H + (size_t)node * HS;
#pragma unroll
    for (int j = 0; j < DH; ++j)
        h[j] = fast_tanh(fmaf(Wi[j * DIN], x0, fmaf(Wi[j * DIN + 1], x1, bi[j])));
    h[DH]     = x0;
    h[DH + 1] = x1;
}

// =====================================================================
// 3) Edge network (WMMA). One wave -> 16 edges.
//    A = We1 in LDS; rows 8..11 zeroed, 12..15 only partly written (unconditional ds loads);
//    B = gathered [H[idx_o] ; H[idx_i]] (K = 20 = 5 chunks of 4);
//    lane n (<16) ends with all 8 hidden pre-acts of edge n in acc[0..7].
// =====================================================================
__global__ void edge_kernel(const float* __restrict__ H,
                            const int* __restrict__ idx_i,
                            const int* __restrict__ idx_o,
                            const float* __restrict__ We1,
                            const float* __restrict__ be1,
                            const float* __restrict__ We2,
                            const float* __restrict__ be2,
                            float* __restrict__ eout) {
    __shared__ float sW1[16 * 20];   // rows 8..11 zero; entries 256..319 never written (256 threads)
    __shared__ float sb1[DH];
    __shared__ float sW2[DH];
    __shared__ float sb2;
    const int t = threadIdx.x;
    if (t < 16 * 20) sW1[t] = (t < DH * 20) ? We1[t] : 0.0f;
    if (t < DH) { sb1[t] = be1[t]; sW2[t] = We2[t]; }
    if (t == 0) sb2 = be2[0];
    __syncthreads();

    const int wave = t >> 5;
    const int lane = t & 31;
    const int row  = lane & 15;   // A: M-row, B: edge column
    const int half = lane >> 4;   // K-pair selector within each 4-chunk

    const int ge = blockIdx.x * 128 + wave * 16 + row;  // global edge
    const int b  = ge >> 11;                            // / E_
    const int no = idx_o[ge];
    const int ni = idx_i[ge];
    const float* Ho = H + (size_t)(b * N_ + no) * HS;
    const float* Hi = H + (size_t)(b * N_ + ni) * HS;

    v8f acc = {};
#pragma unroll
    for (int c = 0; c < 5; ++c) {
        const int k0 = 4 * c + 2 * half;
        const int k1 = k0 + 1;
        v2f a, bm;
        a.x  = sW1[row * 20 + k0];
        a.y  = sW1[row * 20 + k1];
        bm.x = (k0 < DF) ? Ho[k0] : Hi[k0 - DF];
        bm.y = (k1 < DF) ? Ho[k1] : Hi[k1 - DF];
        acc = __builtin_amdgcn_wmma_f32_16x16x4_f32(
                  false, a, false, bm, (short)0, acc, false, false);
    }

    if (half == 0) {
        float z = sb2;
#pragma unroll
        for (int r = 0; r < DH; ++r) {
            float h = fast_tanh(acc[r] + sb1[r]);
            z = fmaf(sW2[r], h, z);
        }
        eout[ge] = fast_sigmoid(z);
    }
}

// =====================================================================
// 4) Prep the per-node message vector M = [0(10) | 0(10) | H(10) | 0,0]
//    (zeroes the scatter targets and stages H so the node kernel reads
//     one contiguous, aligned 32-float vector per node).
// =====================================================================
__global__ void prep_m_kernel(const float* __restrict__ H,
                              float* __restrict__ Mbuf) {
    int i = blockIdx.x * blockDim.x + threadIdx.x;   // over B*N*MS
    int node = i >> 5;
    int k    = i & (MS - 1);
    float v = 0.0f;
    if (k >= 2 * DF && k < 3 * DF) v = H[(size_t)node * HS + (k - 2 * DF)];
    Mbuf[i] = v;
}

// =====================================================================
// 5) Weighted scatter into M: M[idx_i][0:10] += e * H[idx_o];
//                             M[idx_o][10:20] += e * H[idx_i]
// =====================================================================
__global__ void scatter_kernel(const float* __restrict__ H,
                               const float* __restrict__ ebuf,
                               const int* __restrict__ idx_i,
                               const int* __restrict__ idx_o,
                               float* __restrict__ Mbuf) {
    int ge = blockIdx.x * blockDim.x + threadIdx.x;
    if (ge >= B_ * E_) return;
    const int b  = ge >> 11;
    const float ev = ebuf[ge];
    const int ni = idx_i[ge];
    const int no = idx_o[ge];
    const float* Ho = H + (size_t)(b * N_ + no) * HS;
    const float* Hi = H + (size_t)(b * N_ + ni) * HS;
    float* pmi = Mbuf + (size_t)(b * N_ + ni) * MS;        // mi slice
    float* pmo = Mbuf + (size_t)(b * N_ + no) * MS + DF;   // mo slice
#pragma unroll
    for (int f = 0; f < DF; ++f) {
        atomicAdd(&pmi[f], ev * Ho[f]);
        atomicAdd(&pmo[f], ev * Hi[f]);
    }
}

// =====================================================================
// 6) Node network (WMMA). One wave -> 16 nodes.
//    A = Wn1 zero-padded to 16x32 in LDS (unconditional ds loads);
//    B = contiguous 32-float M vector (unconditional b64 loads).
// =====================================================================
__global__ void node_kernel(const float* __restrict__ Mbuf,
                            const float* __restrict__ X,
                            const float* __restrict__ Wn1,
                            const float* __restrict__ bn1,
                            const float* __restrict__ Wn2,
                            const float* __restrict__ bn2,
                            float* __restrict__ Hout) {
    __shared__ float sW1[16 * 32];   // rows 8..15 and cols 30,31 zero
    __shared__ float sb1[DH];
    __shared__ float sW2[DH * DH];
    __shared__ float sb2[DH];
    const int t = threadIdx.x;
    for (int i = t; i < 16 * 32; i += 256) {
        int r = i >> 5, k = i & 31;
        sW1[i] = (r < DH && k < 30) ? Wn1[r * 30 + k] : 0.0f;
    }
    if (t < DH * DH) sW2[t] = Wn2[t];
    if (t < DH) { sb1[t] = bn1[t]; sb2[t] = bn2[t]; }
    __syncthreads();

    const int wave = t >> 5;
    const int lane = t & 31;
    const int row  = lane & 15;
    const int half = lane >> 4;

    const int gn = blockIdx.x * 128 + wave * 16 + row;  // global node
    const float* Fm = Mbuf + (size_t)gn * MS;

    v8f acc = {};
#pragma unroll
    for (int c = 0; c < 8; ++c) {            // K = 30 padded to 32
        const int k0 = 4 * c + 2 * half;
        v2f a, bm;
        a.x  = sW1[row * 32 + k0];
        a.y  = sW1[row * 32 + k0 + 1];
        bm.x = Fm[k0];
        bm.y = Fm[k0 + 1];
        acc = __builtin_amdgcn_wmma_f32_16x16x4_f32(
                  false, a, false, bm, (short)0, acc, false, false);
    }

    if (half == 0) {
        float h[DH];
#pragma unroll
        for (int r = 0; r < DH; ++r) h[r] = fast_tanh(acc[r] + sb1[r]);
        float* out = Hout + (size_t)gn * HS;
#pragma unroll
        for (int j = 0; j < DH; ++j) {
            float z = sb2[j];
#pragma unroll
            for (int r = 0; r < DH; ++r) z = fmaf(sW2[j * DH + r], h[r], z);
            out[j] = fast_tanh(z);
        }
        out[DH]     = X[gn * DIN + 0];
        out[DH + 1] = X[gn * DIN + 1];
    }
}

// =====================================================================
// Host-side orchestration
// =====================================================================
extern "C" void kernel_launch(void* const* d_in, const int* in_sizes, int n_in,
                              void* d_out, int out_size, void* d_ws, size_t ws_size,
                              hipStream_t stream) {
    const float* X     = (const float*)d_in[0];
    const float* Ri    = (const float*)d_in[1];
    const float* Ro    = (const float*)d_in[2];
    const float* Wi    = (const float*)d_in[3];
    const float* bi_in = (const float*)d_in[4];
    const float* We1   = (const float*)d_in[5];
    const float* be1   = (const float*)d_in[6];
    const float* We2   = (const float*)d_in[7];
    const float* be2   = (const float*)d_in[8];
    const float* Wn1   = (const float*)d_in[9];
    const float* bn1   = (const float*)d_in[10];
    const float* Wn2   = (const float*)d_in[11];
    const float* bn2   = (const float*)d_in[12];

    // workspace layout
    int*   idx_i   = (int*)d_ws;                         // NCOLS ints
    int*   idx_o   = idx_i + NCOLS;                      // NCOLS ints
    float* Ha      = (float*)(idx_o + NCOLS);            // B*N*HS
    float* Hb      = Ha + (size_t)B_ * N_ * HS;          // B*N*HS
    float* Mbuf    = Hb + (size_t)B_ * N_ * HS;          // B*N*MS
    float* ebuf    = Mbuf + (size_t)B_ * N_ * MS;        // NCOLS
    float* partial = ebuf + NCOLS;                       // 2*NSPLIT*NCOLS

    // 1) decode both incidence matrices in one wide launch (256 MB pass)
    {
        dim3 grid((NCOLS / 4) / 256, NSPLIT, 2);
        decode_partial_kernel<<<grid, 256, 0, stream>>>(Ri, Ro, partial);
        decode_combine_kernel<<<(2 * NCOLS) / 256, 256, 0, stream>>>(
            partial, idx_i, idx_o);
    }

    // 2) encoder
    encoder_kernel<<<(B_ * N_) / 256, 256, 0, stream>>>(X, Wi, bi_in, Ha);

    // 3) message-passing iterations
    float* Hc = Ha;
    float* Hn = Hb;
    for (int it = 0; it < 3; ++it) {
        edge_kernel<<<(B_ * E_) / 128, 256, 0, stream>>>(
            Hc, idx_i, idx_o, We1, be1, We2, be2, ebuf);
        prep_m_kernel<<<(B_ * N_ * MS) / 256, 256, 0, stream>>>(Hc, Mbuf);
        scatter_kernel<<<(B_ * E_) / 256, 256, 0, stream>>>(
            Hc, ebuf, idx_i, idx_o, Mbuf);
        node_kernel<<<(B_ * N_) / 128, 256, 0, stream>>>(
            Mbuf, X, Wn1, bn1, Wn2, bn2, Hn);
        float* tmp = Hc; Hc = Hn; Hn = tmp;
    }

    // 4) final edge network -> d_out [B,E]
    edge_kernel<<<(B_ * E_) / 128, 256, 0, stream>>>(
        Hc, idx_i, idx_o, We1, be1, We2, be2, (float*)d_out);
}